// GHMC_loss_16535624089725
// MI455X (gfx1250) — compile-verified
//
#include <hip/hip_runtime.h>
#include <hip/hip_bf16.h>

// ---------------------------------------------------------------------------
// GHM-C loss for MI455X (gfx1250, wave32).
//   Pass 1 (grid-wide, memory-bound): stream pred/target/label_weight once
//     (126 MB @ 23.3 TB/s ~ 5.4us floor), per-thread LDS bins via ds_add_f32,
//     counts packed in a u64 (6 bits/bin), block tree-reduce, write a
//     16-row x NB-col matrix of counts and one of sums into d_ws.
//   Pass 2 (1 wave): V_WMMA_F32_16X16X4_F32 x ones reduces the partial
//     matrices (row = bin), then loss = (1/n) * sum_b S_b / cnt_b
//     (the `total` factor cancels algebraically).
// ---------------------------------------------------------------------------

typedef float v2f __attribute__((ext_vector_type(2)));
typedef float v8f __attribute__((ext_vector_type(8)));
typedef float v4f __attribute__((ext_vector_type(4)));
typedef int   v4i __attribute__((ext_vector_type(4)));

#define BINS 10
#define TPB 256
#define BIN_STRIDE 12  // pad 10 -> 12 dwords per thread to spread LDS banks

__device__ __forceinline__ void ghmc_elem(float p, int t, int lw,
                                          float* __restrict__ sum_base,
                                          unsigned long long& packed) {
  bool valid = lw > 0;
  float tv = (float)t;
  // g = |sigmoid(p) - t| = sigmoid(t ? -p : p)
  float q = t ? -p : p;
  float g = __builtin_amdgcn_rcpf(1.0f + __expf(-q));
  int b = (int)(g * 10.0f);
  b = b > 9 ? 9 : b;
  // histogram: route invalid elements to bin 10 (bits 60..63, harmless)
  int bc = valid ? b : 10;
  packed += 1ull << (6 * bc);
  // bce(logits) = softplus(p) - t*p  (stable: max(p,0)+log1p(exp(-|p|)))
  float sp = fmaxf(p, 0.0f) + __logf(1.0f + __expf(-fabsf(p)));
  float bce = sp - tv * p;
  float w = valid ? bce : 0.0f;
  atomicAdd(sum_base + b, w);  // ds_add_f32, thread-private slot: no contention
}

__global__ __launch_bounds__(TPB) void ghmc_pass1(
    const float* __restrict__ pred, const int* __restrict__ target,
    const int* __restrict__ lweight, float* __restrict__ part, int NB, int N) {
  __shared__ float s_sum[TPB * BIN_STRIDE];
  __shared__ float s_cnt[TPB * BIN_STRIDE];
  const int tid = threadIdx.x;

  #pragma unroll
  for (int k = 0; k < BIN_STRIDE; ++k) s_sum[tid * BIN_STRIDE + k] = 0.0f;
  __syncthreads();

  float* my_sum = &s_sum[tid * BIN_STRIDE];
  unsigned long long packed = 0ull;
  float cf[BINS];
  #pragma unroll
  for (int b = 0; b < BINS; ++b) cf[b] = 0.0f;

  const int n4 = N >> 2;
  const v4f* __restrict__ p4 = (const v4f*)pred;
  const v4i* __restrict__ t4 = (const v4i*)target;
  const v4i* __restrict__ w4 = (const v4i*)lweight;

  int since = 0;
  const int stride = gridDim.x * TPB;
  for (int i = blockIdx.x * TPB + tid; i < n4; i += stride) {
    v4f p = __builtin_nontemporal_load(&p4[i]);
    v4i t = __builtin_nontemporal_load(&t4[i]);
    v4i lw = __builtin_nontemporal_load(&w4[i]);
    ghmc_elem(p.x, t.x, lw.x, my_sum, packed);
    ghmc_elem(p.y, t.y, lw.y, my_sum, packed);
    ghmc_elem(p.z, t.z, lw.z, my_sum, packed);
    ghmc_elem(p.w, t.w, lw.w, my_sum, packed);
    if (++since == 15) {  // flush <=60 elems/chunk; 6-bit fields hold <=63
      #pragma unroll
      for (int b = 0; b < BINS; ++b)
        cf[b] += (float)((packed >> (6 * b)) & 63ull);
      packed = 0ull;
      since = 0;
    }
  }
  // scalar tail (N % 4)
  int rem = N & 3;
  if (blockIdx.x == 0 && tid < rem) {
    int i = n4 * 4 + tid;
    ghmc_elem(pred[i], target[i], lweight[i], my_sum, packed);
  }
  #pragma unroll
  for (int b = 0; b < BINS; ++b)
    cf[b] += (float)((packed >> (6 * b)) & 63ull);

  #pragma unroll
  for (int b = 0; b < BINS; ++b) s_cnt[tid * BIN_STRIDE + b] = cf[b];
  __syncthreads();

  // block tree reduction over the thread dimension
  for (int s = TPB / 2; s > 0; s >>= 1) {
    if (tid < s) {
      #pragma unroll
      for (int b = 0; b < BINS; ++b) {
        s_sum[tid * BIN_STRIDE + b] += s_sum[(tid + s) * BIN_STRIDE + b];
        s_cnt[tid * BIN_STRIDE + b] += s_cnt[(tid + s) * BIN_STRIDE + b];
      }
    }
    __syncthreads();
  }

  // partials matrix: rows 0..15 = counts (bins 0..9, rows 10..15 zero),
  // rows 16..31 = sums. Column = block id. Row length = NB.
  if (tid < 16) {
    float cv = (tid < BINS) ? s_cnt[tid] : 0.0f;
    float sv = (tid < BINS) ? s_sum[tid] : 0.0f;
    part[tid * NB + blockIdx.x] = cv;
    part[(16 + tid) * NB + blockIdx.x] = sv;
  }
}

// Single-wave finisher: WMMA (16x16x4 f32) x ones-vector column reduction.
__global__ __launch_bounds__(32) void ghmc_pass2(
    const float* __restrict__ part, int NB, float* __restrict__ out) {
  const int L = threadIdx.x;
  const int m = L & 15;           // A-matrix row (bin) held by this lane
  const int koff = (L >> 4) << 1; // lanes 0-15: K=0,1 ; lanes 16-31: K=2,3
  const float* __restrict__ cntrow = part + m * NB;
  const float* __restrict__ sumrow = part + (16 + m) * NB;

  v8f cc = {};  // count accumulator, D[m][*] = sum over columns of row m
  v8f cs = {};  // sum accumulator
  v2f ones; ones.x = 1.0f; ones.y = 1.0f;

  for (int col = 0; col < NB; col += 4) {
    v2f a;
    a.x = cntrow[col + koff];
    a.y = cntrow[col + koff + 1];
    cc = __builtin_amdgcn_wmma_f32_16x16x4_f32(false, a, false, ones,
                                               (short)0, cc, false, false);
    a.x = sumrow[col + koff];
    a.y = sumrow[col + koff + 1];
    cs = __builtin_amdgcn_wmma_f32_16x16x4_f32(false, a, false, ones,
                                               (short)0, cs, false, false);
  }

  // C/D layout: VGPR r holds rows r (lanes 0-15) and r+8 (lanes 16-31).
  float cnt[BINS], sm[BINS];
  #pragma unroll
  for (int b = 0; b < 8; ++b) {
    cnt[b] = __shfl(cc[b], 0, 32);
    sm[b] = __shfl(cs[b], 0, 32);
  }
  cnt[8] = __shfl(cc[0], 16, 32);
  cnt[9] = __shfl(cc[1], 16, 32);
  sm[8] = __shfl(cs[0], 16, 32);
  sm[9] = __shfl(cs[1], 16, 32);

  // loss = (1/n) * sum_{cnt_b>0} S_b / cnt_b   (total cancels)
  int n = 0;
  float acc = 0.0f;
  #pragma unroll
  for (int b = 0; b < BINS; ++b) {
    if (cnt[b] > 0.0f) {
      n++;
      acc += sm[b] / cnt[b];
    }
  }
  float loss = (n > 0) ? acc / (float)n : 0.0f;
  if (L == 0) out[0] = loss;
}

extern "C" void kernel_launch(void* const* d_in, const int* in_sizes, int n_in,
                              void* d_out, int out_size, void* d_ws, size_t ws_size,
                              hipStream_t stream) {
  const float* pred = (const float*)d_in[0];
  const int* target = (const int*)d_in[1];
  const int* lweight = (const int*)d_in[2];
  const int N = in_sizes[0];

  int NB = 1024;  // 8192 wave32s; partial matrix = 32 rows x NB cols = 128 KB
  size_t need = (size_t)32 * (size_t)NB * sizeof(float);
  if (ws_size < need) {
    NB = (int)((ws_size / (32 * sizeof(float))) & ~(size_t)3);
    if (NB < 4) NB = 4;
  }
  float* part = (float*)d_ws;

  ghmc_pass1<<<NB, TPB, 0, stream>>>(pred, target, lweight, part, NB, N);
  ghmc_pass2<<<1, 32, 0, stream>>>(part, NB, (float*)d_out);
}